// GINEModelWithVirtualNode_57062935495525
// MI455X (gfx1250) — compile-verified
//
#include <hip/hip_runtime.h>
#include <hip/hip_bf16.h>

typedef float v2f __attribute__((ext_vector_type(2)));
typedef float v8f __attribute__((ext_vector_type(8)));

#define EMB   300
#define H2    600
#define C1    150
#define NCLS  6
#define GNUM  128
#define EDGED 7
#define BN_EPSF 1e-5f

// ---------------------------------------------------------------- elementwise

__global__ void k_zero(float* __restrict__ p, long n) {
  long i = blockIdx.x * (long)blockDim.x + threadIdx.x;
  if (i < n) p[i] = 0.0f;
}

__global__ void k_gather_embed(const float* __restrict__ emb, const int* __restrict__ x,
                               float* __restrict__ h, int N) {
  long i = blockIdx.x * (long)blockDim.x + threadIdx.x;
  long n = i / EMB;
  int  c = (int)(i - n * EMB);
  if (n < N) h[i] = emb[(long)x[n] * EMB + c];
}

__global__ void k_init_vn(const float* __restrict__ vne, float* __restrict__ vn) {
  int i = blockIdx.x * blockDim.x + threadIdx.x;
  if (i < GNUM * EMB) vn[i] = vne[i % EMB];
}

__global__ void k_add_vn(float* __restrict__ h, const float* __restrict__ vn,
                         const int* __restrict__ batch, int N) {
  long i = blockIdx.x * (long)blockDim.x + threadIdx.x;
  long n = i / EMB;
  int  c = (int)(i - n * EMB);
  if (n < N) h[i] += vn[(long)batch[n] * EMB + c];
}

// zb = (1+eps)*h + zb   (zb holds aggr on input)
__global__ void k_zcombine(float* __restrict__ zb, const float* __restrict__ h,
                           const float* __restrict__ eps, long n) {
  long i = blockIdx.x * (long)blockDim.x + threadIdx.x;
  if (i < n) zb[i] = (1.0f + eps[0]) * h[i] + zb[i];
}

__global__ void k_segsum(const float* __restrict__ h, const int* __restrict__ batch,
                         float* __restrict__ out, int N) {
  long i = blockIdx.x * (long)blockDim.x + threadIdx.x;
  long n = i / EMB;
  int  c = (int)(i - n * EMB);
  if (n < N) atomicAdd(&out[(long)batch[n] * EMB + c], h[i]);
}

__global__ void k_add2(const float* __restrict__ a, const float* __restrict__ b,
                       float* __restrict__ o, long n) {
  long i = blockIdx.x * (long)blockDim.x + threadIdx.x;
  if (i < n) o[i] = a[i] + b[i];
}

__global__ void k_accum(float* __restrict__ a, const float* __restrict__ t, long n) {
  long i = blockIdx.x * (long)blockDim.x + threadIdx.x;
  if (i < n) a[i] += t[i];
}

// ------------------------------------------- fused edge message + aggregation
// One wave (32 lanes) per edge; edge_W (7x300) + edge_b staged in LDS.
// msg = relu(h[src] + edge_attr[e] @ edge_W + edge_b); atomic scatter to aggr[dst].
// Avoids materializing e = edge_attr @ edge_W ([E,300] = 312 MB) entirely.
__global__ void k_edge_msg(const int* __restrict__ ei, const float* __restrict__ eattr,
                           const float* __restrict__ h, const float* __restrict__ eW,
                           const float* __restrict__ eB, float* __restrict__ aggr, int E) {
  __shared__ float sW[(EDGED + 1) * EMB];  // 7 weight rows + bias row
  for (int i = threadIdx.x; i < (EDGED + 1) * EMB; i += blockDim.x)
    sW[i] = (i < EDGED * EMB) ? eW[i] : eB[i - EDGED * EMB];
  __syncthreads();

  int lane    = threadIdx.x & 31;
  int wid     = blockIdx.x * (blockDim.x >> 5) + (threadIdx.x >> 5);
  int wstride = gridDim.x * (blockDim.x >> 5);

  for (int e = wid; e < E; e += wstride) {
    int src = ei[e];
    int dst = ei[E + e];
    float ea[EDGED];
#pragma unroll
    for (int k = 0; k < EDGED; ++k) ea[k] = eattr[(long)e * EDGED + k];
    const float* hs = h + (long)src * EMB;
    float*       ag = aggr + (long)dst * EMB;
    for (int c = lane; c < EMB; c += 32) {
      float v = hs[c] + sW[EDGED * EMB + c];
#pragma unroll
      for (int k = 0; k < EDGED; ++k) v = fmaf(ea[k], sW[k * EMB + c], v);
      atomicAdd(&ag[c], fmaxf(v, 0.0f));
    }
  }
}

// ------------------------------------------------------------- WMMA f32 GEMM
// out[M,NC] = A[M,K] @ W[K,NC] + bias (+optional ReLU).   REQUIRES K % 4 == 0.
// One 16x16 output tile per wave, V_WMMA_F32_16X16X4_F32, K stepped by 4.
// A-frag (16x4): lanes 0-15 carry M=0..15 / K={k,k+1}; lanes 16-31 K={k+2,k+3}.
// B/C/D row-striped across lanes (C/D VGPR r -> M=r lanes 0-15, M=r+8 lanes 16-31).
// Edge tiles use CLAMPED indices (duplicate row/col, never stored) so the inner
// loop has zero predication: one b64 load + two b32 loads + one WMMA per step.
__global__ void k_gemm_wmma(const float* __restrict__ A, int lda,
                            const float* __restrict__ W, int ldw,
                            const float* __restrict__ bias,
                            float* __restrict__ out, int ldo,
                            int M, int K, int NC, int act) {
  int wave = blockIdx.x * (blockDim.x >> 5) + (threadIdx.x >> 5);
  int lane = threadIdx.x & 31;
  int nTiles = (NC + 15) >> 4;
  int mTiles = (M + 15) >> 4;
  if (wave >= mTiles * nTiles) return;  // wave-uniform exit: EXEC all-1s at WMMA

  int mT = wave / nTiles, nT = wave - mT * nTiles;
  int m0 = mT << 4, n0 = nT << 4;
  int half = lane >> 4;   // 0: K pair {k,k+1}; 1: {k+2,k+3}
  int l15  = lane & 15;

  int arow = m0 + l15; if (arow >= M) arow = M - 1;       // clamp (never stored)
  int bcol = n0 + l15;
  bool bok = bcol < NC;
  int bcolc = bok ? bcol : (NC - 1);                      // clamp (never stored)

  const float* aP = A + (size_t)arow * lda + (half << 1);       // +4 per step
  const float* bP = W + (size_t)(half << 1) * ldw + bcolc;      // +4*ldw per step

  v8f acc = {};
  for (int k = 0; k < K; k += 4) {
    v2f a = *(const v2f*)aP;              // global_load_b64 (8B-aligned: lda%2==0)
    v2f b;
    b.x = bP[0];
    b.y = bP[ldw];
    acc = __builtin_amdgcn_wmma_f32_16x16x4_f32(false, a, false, b,
                                                (short)0, acc, false, false);
    aP += 4;
    bP += (size_t)(ldw << 2);
  }

  float bv = bias[bcolc];
#pragma unroll
  for (int r = 0; r < 8; ++r) {
    int row = m0 + r + (half << 3);
    float v = acc[r] + bv;
    if (act) v = fmaxf(v, 0.0f);
    if (row < M && bok) out[(size_t)row * ldo + bcol] = v;
  }
}

// ------------------------------------------------ tiny classifier head (K=150)
// logits[g][c] = sum_k Q[g][k] * Wc2[k][c] + bc2[c]   (128x150x6: 0.2 MFLOP)
__global__ void k_head(const float* __restrict__ Q, const float* __restrict__ Wc2,
                       const float* __restrict__ bc2, float* __restrict__ logits) {
  int i = blockIdx.x * blockDim.x + threadIdx.x;
  if (i >= GNUM * NCLS) return;
  int g = i / NCLS, c = i - g * NCLS;
  float s = bc2[c];
  const float* q = Q + (size_t)g * C1;
  for (int k = 0; k < C1; ++k) s = fmaf(q[k], Wc2[(size_t)k * NCLS + c], s);
  logits[i] = s;
}

// ------------------------------------------------- BatchNorm (batch statistics)
// One block handles 32 consecutive columns (coalesced); 8 row-groups LDS-reduced.
__global__ void k_bn_stats(const float* __restrict__ y, int M, int C,
                           float* __restrict__ mu, float* __restrict__ var) {
  int colLocal = threadIdx.x & 31;
  int rgrp     = threadIdx.x >> 5;     // 0..7
  int col      = blockIdx.x * 32 + colLocal;
  float s1 = 0.0f, s2 = 0.0f;
  if (col < C) {
    for (int r = rgrp; r < M; r += 8) {
      float v = y[(long)r * C + col];
      s1 += v; s2 += v * v;
    }
  }
  __shared__ float sh1[8][32], sh2[8][32];
  sh1[rgrp][colLocal] = s1;
  sh2[rgrp][colLocal] = s2;
  __syncthreads();
  if (rgrp == 0) {
#pragma unroll
    for (int g = 1; g < 8; ++g) { s1 += sh1[g][colLocal]; s2 += sh2[g][colLocal]; }
    if (col < C) {
      float m = s1 / (float)M;
      mu[col]  = m;
      var[col] = fmaxf(s2 / (float)M - m * m, 0.0f);
    }
  }
}

__global__ void k_bn_apply_relu(float* __restrict__ y, long n, int C,
                                const float* __restrict__ mu, const float* __restrict__ var,
                                const float* __restrict__ g, const float* __restrict__ b) {
  long i = blockIdx.x * (long)blockDim.x + threadIdx.x;
  if (i >= n) return;
  int c = (int)(i % C);
  float v = y[i];
  v = g[c] * (v - mu[c]) * rsqrtf(var[c] + BN_EPSF) + b[c];
  y[i] = fmaxf(v, 0.0f);
}

// ---------------------------------------------------------------------- host

extern "C" void kernel_launch(void* const* d_in, const int* in_sizes, int n_in,
                              void* d_out, int out_size, void* d_ws, size_t ws_size,
                              hipStream_t stream) {
  const int N = in_sizes[0];
  const int E = in_sizes[1] / 2;

  const int*   x        = (const int*)d_in[0];
  const int*   ei       = (const int*)d_in[1];
  const float* eattr    = (const float*)d_in[2];
  const int*   batch    = (const int*)d_in[3];
  // d_in[4] = num_graphs scalar (G fixed at 128 by the reference setup)
  const float* node_emb = (const float*)d_in[5];
  const float* edgeW    = (const float*)d_in[6];
  const float* edgeB    = (const float*)d_in[7];
  const float* vn_emb   = (const float*)d_in[8];
  const int LBASE = 9;                 // 5 layers x 9 tensors
  const int VBASE = LBASE + 5 * 9;     // 4 vn-mlps x 8 tensors
  const int CBASE = VBASE + 4 * 8;     // Wc1, bc1, Wc2, bc2
  const float* Wc1 = (const float*)d_in[CBASE + 0];
  const float* bc1 = (const float*)d_in[CBASE + 1];
  const float* Wc2 = (const float*)d_in[CBASE + 2];
  const float* bc2 = (const float*)d_in[CBASE + 3];

  // workspace carve-up
  float* ws = (float*)d_ws;
  const long NE = (long)N * EMB;
  float* bufA   = ws;                       ws += NE;
  float* bufB   = ws;                       ws += NE;
  float* vn     = ws;                       ws += (long)GNUM * EMB;
  float* pooled = ws;                       ws += (long)GNUM * EMB;
  float* t1     = ws;                       ws += (long)GNUM * EMB;
  float* t2     = ws;                       ws += (long)GNUM * H2;
  float* Q      = ws;                       ws += (long)GNUM * C1;
  float* mu     = ws;                       ws += H2;
  float* var    = ws;                       ws += H2;

  auto cdiv = [](long a, long b) { return (int)((a + b - 1) / b); };

  auto gemm = [&](const float* Am, const float* Wm, const float* bm, float* om,
                  int M, int K, int NC, int act) {
    int tiles = ((M + 15) >> 4) * ((NC + 15) >> 4);
    k_gemm_wmma<<<cdiv(tiles, 8), 256, 0, stream>>>(Am, K, Wm, NC, bm, om, NC,
                                                    M, K, NC, act);
  };
  auto bn = [&](float* y, int M, int C, const float* g, const float* b) {
    k_bn_stats<<<cdiv(C, 32), 256, 0, stream>>>(y, M, C, mu, var);
    long n = (long)M * C;
    k_bn_apply_relu<<<cdiv(n, 256), 256, 0, stream>>>(y, n, C, mu, var, g, b);
  };

  // h = node_emb[x]; vn = broadcast(vn_emb)
  k_gather_embed<<<cdiv(NE, 256), 256, 0, stream>>>(node_emb, x, bufA, N);
  k_init_vn<<<cdiv((long)GNUM * EMB, 256), 256, 0, stream>>>(vn_emb, vn);

  float* h     = bufA;
  float* other = bufB;

  for (int l = 0; l < 5; ++l) {
    const float* eps   = (const float*)d_in[LBASE + 9 * l + 0];
    const float* W1    = (const float*)d_in[LBASE + 9 * l + 1];
    const float* b1    = (const float*)d_in[LBASE + 9 * l + 2];
    const float* bn1g  = (const float*)d_in[LBASE + 9 * l + 3];
    const float* bn1b  = (const float*)d_in[LBASE + 9 * l + 4];
    const float* W2    = (const float*)d_in[LBASE + 9 * l + 5];
    const float* b2    = (const float*)d_in[LBASE + 9 * l + 6];
    const float* bnog  = (const float*)d_in[LBASE + 9 * l + 7];
    const float* bnob  = (const float*)d_in[LBASE + 9 * l + 8];

    // h += vn[batch]
    k_add_vn<<<cdiv(NE, 256), 256, 0, stream>>>(h, vn, batch, N);
    // aggr (in `other`) = sum over edges of relu(h[src] + edge_attr@edge_W + edge_b)
    k_zero<<<cdiv(NE, 256), 256, 0, stream>>>(other, NE);
    {
      int blocks = cdiv(E, 8);  // 8 waves/block, 1 edge per wave per pass
      if (blocks > 2048) blocks = 2048;
      k_edge_msg<<<blocks, 256, 0, stream>>>(ei, eattr, h, edgeW, edgeB, other, E);
    }
    // z = (1+eps)*h + aggr   (into `other`)
    k_zcombine<<<cdiv(NE, 256), 256, 0, stream>>>(other, h, eps, NE);
    // MLP: BN(relu) o Linear, twice
    gemm(other, W1, b1, h, N, EMB, EMB, 0);
    bn(h, N, EMB, bn1g, bn1b);
    gemm(h, W2, b2, other, N, EMB, EMB, 0);
    bn(other, N, EMB, bnog, bnob);
    { float* tmp = h; h = other; other = tmp; }  // new h

    if (l < 4) {
      const float* Wa   = (const float*)d_in[VBASE + 8 * l + 0];
      const float* ba   = (const float*)d_in[VBASE + 8 * l + 1];
      const float* bnag = (const float*)d_in[VBASE + 8 * l + 2];
      const float* bnab = (const float*)d_in[VBASE + 8 * l + 3];
      const float* Wb   = (const float*)d_in[VBASE + 8 * l + 4];
      const float* bb   = (const float*)d_in[VBASE + 8 * l + 5];
      const float* bnbg = (const float*)d_in[VBASE + 8 * l + 6];
      const float* bnbb = (const float*)d_in[VBASE + 8 * l + 7];

      long GE = (long)GNUM * EMB;
      k_zero<<<cdiv(GE, 256), 256, 0, stream>>>(pooled, GE);
      k_segsum<<<cdiv(NE, 256), 256, 0, stream>>>(h, batch, pooled, N);
      k_add2<<<cdiv(GE, 256), 256, 0, stream>>>(vn, pooled, t1, GE);
      gemm(t1, Wa, ba, t2, GNUM, EMB, H2, 0);
      bn(t2, GNUM, H2, bnag, bnab);
      gemm(t2, Wb, bb, t1, GNUM, H2, EMB, 0);
      bn(t1, GNUM, EMB, bnbg, bnbb);
      k_accum<<<cdiv(GE, 256), 256, 0, stream>>>(vn, t1, GE);
    }
  }

  // outputs: [logits (128*6) | graph_emb (128*300)]
  float* logits    = (float*)d_out;
  float* graph_emb = logits + (long)GNUM * NCLS;
  long GE = (long)GNUM * EMB;
  k_zero<<<cdiv(GE, 256), 256, 0, stream>>>(graph_emb, GE);
  k_segsum<<<cdiv(NE, 256), 256, 0, stream>>>(h, batch, graph_emb, N);
  gemm(graph_emb, Wc1, bc1, Q, GNUM, EMB, C1, 1);   // + ReLU (WMMA, K=300)
  k_head<<<cdiv((long)GNUM * NCLS, 256), 256, 0, stream>>>(Q, Wc2, bc2, logits);
}